// ConvZ3P24_63359357551423
// MI455X (gfx1250) — compile-verified
//
#include <hip/hip_runtime.h>
#include <hip/hip_bf16.h>

typedef __attribute__((ext_vector_type(2))) float v2f;
typedef __attribute__((ext_vector_type(4))) float v4f;
typedef __attribute__((ext_vector_type(8))) float v8f;

#define KTOT 108       // 27 taps * CIN(4), tap-major: k' = tap*4 + c
#define NCH  192       // N_ROT(24) * COUT(8)
#define DIM  64
#define SPATIAL (DIM*DIM*DIM)      // 262144 = 1<<18
#define NPAIR (KTOT/2)             // 54 k-row pairs
#define WROW  (NCH*2)              // 384 floats per pair-row
#define PD   66                    // padded dim
#define PD2  (PD*PD)               // 4356
#define PSPATIAL (PD*PD*PD)        // 287496
#define XPAD_OFF_FLOATS 32768      // X_pad at +128KB in workspace

// ---------------------------------------------------------------------------
// Rotation matrix r (0..23), same ordering as the Python reference.
// ---------------------------------------------------------------------------
__device__ inline void rot_mat(int r, int R[3][3]) {
    const int perms[6][3] = {{0,1,2},{0,2,1},{1,0,2},{1,2,0},{2,0,1},{2,1,0}};
    const int par[6]      = { 1,     -1,     -1,      1,      1,     -1    };
    int cnt = 0;
    for (int p = 0; p < 6; ++p) {
        for (int s = 0; s < 8; ++s) {
            int sg0 = (s & 4) ? -1 : 1;
            int sg1 = (s & 2) ? -1 : 1;
            int sg2 = (s & 1) ? -1 : 1;
            if (par[p] * sg0 * sg1 * sg2 > 0) {
                if (cnt == r) {
                    for (int i = 0; i < 3; ++i)
                        for (int j = 0; j < 3; ++j) R[i][j] = 0;
                    R[0][perms[p][0]] = sg0;
                    R[1][perms[p][1]] = sg1;
                    R[2][perms[p][2]] = sg2;
                    return;
                }
                ++cnt;
            }
        }
    }
}

// ---------------------------------------------------------------------------
// Kernel 1: rotated weight matrix, tap-major K, b128-friendly layout:
//   (k', j) -> ws[(k'>>1)*WROW + (j&15)*24 + (j>>4)*2 + (k'&1)]
// ---------------------------------------------------------------------------
__global__ void build_wmat_kernel(const float* __restrict__ w,
                                  float* __restrict__ wmat) {
    int e = blockIdx.x * blockDim.x + threadIdx.x;
    if (e >= KTOT * NCH) return;
    int kp = e / NCH, j = e - kp * NCH;      // kp = k' (tap-major)
    int tap = kp >> 2, c = kp & 3;
    int kd = tap / 9, kh = (tap - kd * 9) / 3, kw = tap % 3;
    int r = j >> 3, o = j & 7;

    int R[3][3];
    rot_mat(r, R);
    int ux = kw - 1, uy = kh - 1, uz = kd - 1;
    int gx = R[0][0]*ux + R[0][1]*uy + R[0][2]*uz;
    int gy = R[1][0]*ux + R[1][1]*uy + R[1][2]*uz;
    int gz = R[2][0]*ux + R[2][1]*uy + R[2][2]*uz;
    int sd = gz + 1, sh = gy + 1, sw = gx + 1;

    float val = w[(((o * 4 + c) * 3 + sd) * 3 + sh) * 3 + sw];
    wmat[(kp >> 1) * WROW + (j & 15) * 24 + (j >> 4) * 2 + (kp & 1)] = val;
}

// ---------------------------------------------------------------------------
// Kernel 1b: zero-padded, CHANNEL-INTERLEAVED input: Xp[n][66][66][66][4].
// One thread per padded spatial point: 4 plane-coalesced loads, 1 b128 store.
// ---------------------------------------------------------------------------
__global__ void pad_kernel(const float* __restrict__ X,
                           float* __restrict__ Xp) {
    unsigned i = blockIdx.x * blockDim.x + threadIdx.x;   // over 2*PSPATIAL
    if (i >= 2u * PSPATIAL) return;
    unsigned nn = i / PSPATIAL, rem = i - nn * PSPATIAL;
    unsigned z = rem / PD2;  rem -= z * PD2;
    unsigned y = rem / PD;
    unsigned x = rem - y * PD;
    v4f v = {};
    if (z - 1u < (unsigned)DIM && y - 1u < (unsigned)DIM &&
        x - 1u < (unsigned)DIM) {
        unsigned s = (z - 1u) * (DIM * DIM) + (y - 1u) * DIM + (x - 1u);
#pragma unroll
        for (int c = 0; c < 4; ++c)
            v[c] = X[(nn * 4 + c) * SPATIAL + s];
    }
    *(v4f*)(Xp + (size_t)i * 4) = v;                      // 16B aligned
}

// ---------------------------------------------------------------------------
// Kernel 2: implicit GEMM via V_WMMA_F32_16X16X4_F32, dual x-subtile.
//   D[16 chan x 16 x] = bias + A[16 chan x K] * B[K x 16 x]   (x2 subtiles)
// Channel-interleaved padded input: each B fragment = ONE global_load_b64
// at an immediate offset from the per-lane base. A fragments feed 4 WMMAs.
// ---------------------------------------------------------------------------
__global__ __launch_bounds__(256) void conv_wmma_kernel(
    const float* __restrict__ Xp, const float* __restrict__ wmat,
    const float* __restrict__ bias, float* __restrict__ Y) {
    extern __shared__ float smem[];
    float* sW = smem;                   // NPAIR * WROW floats
    float* sB = smem + NPAIR * WROW;    // 8 floats

    unsigned tid = threadIdx.x;
    for (unsigned i = tid; i < NPAIR * WROW; i += 256) sW[i] = wmat[i];
    if (tid < 8) sB[tid] = bias[tid];
    __syncthreads();

    const int wave  = tid >> 5;
    const int lane  = tid & 31;
    const int laneN = lane & 15;
    const int laneH = lane >> 4;

    const int tile = blockIdx.x * 8 + wave;          // 0 .. 16383
    const int x0 = (tile & 1) << 5;                  // 32-wide x strip
    const int yy = (tile >> 1) & 63;
    const int zz = (tile >> 7) & 63;
    const int n  = tile >> 13;

    v8f binit;
#pragma unroll
    for (int g = 0; g < 8; ++g) binit[g] = sB[g];
    v8f accL[12], accR[12];

    // Interleaved-layout base: ((n*PSPATIAL + spatial) * 4) + channel pair.
    const unsigned base0 =
        ((unsigned)n * (unsigned)PSPATIAL +
         (unsigned)(zz * PD2 + yy * PD + x0 + laneN)) * 4u +
        2u * (unsigned)laneH;
    const float* wlane = sW + laneN * 24 + laneH * WROW;

    // ---- peeled K-step 0: C operand = shared bias vector ----
    {
        v2f bL = *(const v2f*)(Xp + base0);          // one b64 load
        v2f bR = *(const v2f*)(Xp + base0 + 64);     // +16 x-positions
        v4f wq[6];
#pragma unroll
        for (int q = 0; q < 6; ++q)
            wq[q] = *(const v4f*)(wlane + q * 4);
#pragma unroll
        for (int q = 0; q < 6; ++q) {
            v2f a0 = __builtin_shufflevector(wq[q], wq[q], 0, 1);
            v2f a1 = __builtin_shufflevector(wq[q], wq[q], 2, 3);
            accL[2*q] = __builtin_amdgcn_wmma_f32_16x16x4_f32(
                false, a0, false, bL, (short)0, binit, false, false);
            accL[2*q+1] = __builtin_amdgcn_wmma_f32_16x16x4_f32(
                false, a1, false, bL, (short)0, binit, false, false);
            accR[2*q] = __builtin_amdgcn_wmma_f32_16x16x4_f32(
                false, a0, false, bR, (short)0, binit, false, false);
            accR[2*q+1] = __builtin_amdgcn_wmma_f32_16x16x4_f32(
                false, a1, false, bR, (short)0, binit, false, false);
        }
    }

    // ---- K-steps 1..26 (tap = ks = kd*9 + kh*3 + kw) ----
#pragma unroll
    for (int ks = 1; ks < 27; ++ks) {
        const int kd = ks / 9;
        const int kh = (ks - kd * 9) / 3;
        const int kw = ks - kd * 9 - kh * 3;
        const unsigned toff = (unsigned)(kd * PD2 + kh * PD + kw) * 4u;
        v2f bL = *(const v2f*)(Xp + base0 + toff);
        v2f bR = *(const v2f*)(Xp + base0 + toff + 64);
        const float* wrow = wlane + (2 * ks) * WROW;
        v4f wq[6];
#pragma unroll
        for (int q = 0; q < 6; ++q)                  // 6x ds_load_b128
            wq[q] = *(const v4f*)(wrow + q * 4);
#pragma unroll
        for (int q = 0; q < 6; ++q) {
            v2f a0 = __builtin_shufflevector(wq[q], wq[q], 0, 1);
            v2f a1 = __builtin_shufflevector(wq[q], wq[q], 2, 3);
            accL[2*q] = __builtin_amdgcn_wmma_f32_16x16x4_f32(
                false, a0, false, bL, (short)0, accL[2*q], false, false);
            accL[2*q+1] = __builtin_amdgcn_wmma_f32_16x16x4_f32(
                false, a1, false, bL, (short)0, accL[2*q+1], false, false);
            accR[2*q] = __builtin_amdgcn_wmma_f32_16x16x4_f32(
                false, a0, false, bR, (short)0, accR[2*q], false, false);
            accR[2*q+1] = __builtin_amdgcn_wmma_f32_16x16x4_f32(
                false, a1, false, bR, (short)0, accR[2*q+1], false, false);
        }
    }

    // Epilogue: immediate per-g offsets (g<<20 B) and +64 B for subtile R.
    const unsigned sp =
        (unsigned)(zz * (DIM * DIM) + yy * DIM + x0 + laneN);
    unsigned yoff = ((unsigned)(n * NCH + 8 * laneH) << 18) + sp;
#pragma unroll
    for (int m = 0; m < 12; ++m) {
#pragma unroll
        for (int g = 0; g < 8; ++g) {
            Y[yoff + ((unsigned)g << 18)] = accL[m][g];
            Y[yoff + ((unsigned)g << 18) + 16] = accR[m][g];
        }
        yoff += 16u << 18;
    }
}

extern "C" void kernel_launch(void* const* d_in, const int* in_sizes, int n_in,
                              void* d_out, int out_size, void* d_ws, size_t ws_size,
                              hipStream_t stream) {
    const float* x      = (const float*)d_in[0];  // (2,4,64,64,64)
    const float* weight = (const float*)d_in[1];  // (8,4,3,3,3)
    const float* bias   = (const float*)d_in[2];  // (8,)
    float* out  = (float*)d_out;                  // (2,24,8,64,64,64)
    float* wmat = (float*)d_ws;                   // [0, 83KB)
    float* xpad = (float*)d_ws + XPAD_OFF_FLOATS; // [128KB, +9.2MB)

    (void)in_sizes; (void)n_in; (void)out_size; (void)ws_size;

    build_wmat_kernel<<<(KTOT * NCH + 255) / 256, 256, 0, stream>>>(weight, wmat);
    pad_kernel<<<(2u * PSPATIAL + 255) / 256, 256, 0, stream>>>(x, xpad);

    size_t lds_bytes = (size_t)(NPAIR * WROW + 8) * sizeof(float);
    conv_wmma_kernel<<<2048, 256, lds_bytes, stream>>>(xpad, wmat, bias, out);
}